// Attention_24257975288003
// MI455X (gfx1250) — compile-verified
//
#include <hip/hip_runtime.h>
#include <cstdint>

#define B_ 16
#define H_ 16
#define S_ 4096
#define D_ 128
#define NSPLIT 16
#define CHUNK (S_ / NSPLIT)   // 256 seq positions per wave
#define TILES (CHUNK / 16)    // 16 tiles of 16 rows
#define LDK 132               // padded LDS row stride in floats (TDM pads 4 DW per 128 DW)
#define MASK_VALUE (-2.3819765e38f)          // -0.7 * FLT_MAX
#define RSQRT_D 0.08838834764831845f         // 1/sqrt(128)

typedef float v2f __attribute__((ext_vector_type(2)));
typedef float v8f __attribute__((ext_vector_type(8)));
typedef unsigned int u32x4 __attribute__((ext_vector_type(4)));
typedef int i32x4 __attribute__((ext_vector_type(4)));
typedef int i32x8 __attribute__((ext_vector_type(8)));

// ---------------------------------------------------------------------------
// Issue one TDM transfer: 16x128 fp32 tile (contiguous 8KB in global) -> LDS,
// with 4 DWORDs of padding inserted after every 128 DWORDs so LDS rows land
// at a 132-float stride (bank-conflict-free for the WMMA fragment reads).
// D# layout per CDNA5 ISA 8.3/8.4. Tracked by TENSORcnt, in-order per wave.
// ---------------------------------------------------------------------------
__device__ __forceinline__ void tdm_load_tile(const float* gsrc, unsigned lds_off) {
  unsigned long long ga = (unsigned long long)(const void*)gsrc;
  u32x4 g0;
  g0[0] = 1u;                                   // count=1 (valid), user mode
  g0[1] = lds_off;                              // LDS byte address
  g0[2] = (unsigned)ga;                         // global_addr[31:0]
  g0[3] = (unsigned)(ga >> 32) | (2u << 30);    // global_addr[56:32] | type=2
  i32x8 g1;
  g1[0] = (2 << 16)        // data_size = 4B
        | (1 << 20)        // pad_enable
        | (6 << 22)        // pad_interval: 128 DWORDs
        | (3 << 25);       // pad_amount:   4 DWORDs  -> row stride 132 floats
  g1[1] = (int)(128u << 16);                    // tensor_dim0 = 128 (bits 79:48)
  g1[2] = (int)(16u << 16);                     // tensor_dim1 = 16  (bits 111:80)
  g1[3] = (int)(128u << 16);                    // tile_dim0  = 128  (bits 127:112)
  g1[4] = 16;                                   // tile_dim1  = 16, tile_dim2 = 0
  g1[5] = 128;                                  // tensor_dim0_stride = 128
  g1[6] = 0;
  g1[7] = 0;
  i32x4 gz4 = {0, 0, 0, 0};                     // groups 2/3 unused (<=2D tile)
  i32x8 gz8 = {0, 0, 0, 0, 0, 0, 0, 0};        // 5th descriptor group unused
  __builtin_amdgcn_tensor_load_to_lds(g0, g1, gz4, gz4, gz8, 0);
}

// ---------------------------------------------------------------------------
// Pass 1: one wave per (b, h, S-chunk). TDM double-buffers K/V tiles into LDS
// while the wave runs V_WMMA_F32_16X16X4_F32 score/PV math + online softmax.
// ---------------------------------------------------------------------------
__global__ __launch_bounds__(32) void attn_partial(
    const float* __restrict__ q, const float* __restrict__ k,
    const float* __restrict__ v,
    const float* __restrict__ k_scaler, const float* __restrict__ v_scaler,
    const int* __restrict__ start, const int* __restrict__ end,
    float* __restrict__ ws_acc, float* __restrict__ ws_m,
    float* __restrict__ ws_l)
{
  __shared__ float lK[2][16 * LDK];
  __shared__ float lV[2][16 * LDK];
  __shared__ float lP[16];

  const int lane = threadIdx.x & 31;
  const int half = lane >> 4;          // 0: lanes 0-15, 1: lanes 16-31
  const int l16  = lane & 15;

  const int item  = blockIdx.x;               // (b*H + h)*NSPLIT + chunk
  const int chunk = item & (NSPLIT - 1);
  const int bh    = item >> 4;                // b*H + h
  const int b     = bh >> 4;

  const int s_base = chunk * CHUNK;
  const int st = start[b];
  const int en = end[b];
  const bool wrap = st > en;

  // LDS byte offsets for the TDM descriptors (low 32 bits of a generic
  // shared-space pointer are the wave-relative LDS address).
  const unsigned offK0 = (unsigned)(size_t)(void*)&lK[0][0];
  const unsigned offK1 = (unsigned)(size_t)(void*)&lK[1][0];
  const unsigned offV0 = (unsigned)(size_t)(void*)&lV[0][0];
  const unsigned offV1 = (unsigned)(size_t)(void*)&lV[1][0];

  const v8f vzero = {0.f, 0.f, 0.f, 0.f, 0.f, 0.f, 0.f, 0.f};

  // q fragments for the QK WMMA B-operand: this half needs q[4c+2*half .. +1]
  const float* qp = q + (size_t)bh * D_ + 2 * half;
  v2f qv[32];
#pragma unroll
  for (int c = 0; c < 32; ++c) qv[c] = *(const v2f*)(qp + 4 * c);

  const float* kbase = k + (size_t)bh * S_ * D_;
  const float* vbase = v + (size_t)bh * S_ * D_;

  float m_run = -__builtin_inff();
  float l_run = 0.0f;
  v8f acc_o[8];
#pragma unroll
  for (int j = 0; j < 8; ++j) acc_o[j] = vzero;

  // Prologue: start DMA of tile 0 into buffer 0.
  tdm_load_tile(kbase + (size_t)s_base * D_, offK0);
  tdm_load_tile(vbase + (size_t)s_base * D_, offV0);
  asm volatile("" ::: "memory");

  for (int t = 0; t < TILES; ++t) {
    const int s0 = s_base + t * 16;
    const int buf = t & 1;

    // Kick off DMA for the next tile into the other buffer, then wait for the
    // current tile. TENSORcnt is in-order per wave: <=2 outstanding ops left
    // means everything issued before them (the current tile) has landed.
    if (t + 1 < TILES) {
      const size_t snext = (size_t)(s0 + 16) * D_;
      tdm_load_tile(kbase + snext, buf ? offK0 : offK1);
      tdm_load_tile(vbase + snext, buf ? offV0 : offV1);
      __builtin_amdgcn_s_wait_tensorcnt(2);
    } else {
      __builtin_amdgcn_s_wait_tensorcnt(0);
    }
    asm volatile("" ::: "memory");

    // ---- scores = K_tile(16xD) . q  via 32 chained f32 WMMAs (K=4 each) ----
    // A: lane<16 holds (row l16, d0..d0+1), lane>=16 holds (row l16, d0+2..3)
    // B: q broadcast across all 16 output columns -> D[s][*] replicated.
    v8f accs = vzero;
#pragma unroll
    for (int c = 0; c < 32; ++c) {
      v2f a = *(const v2f*)&lK[buf][l16 * LDK + 4 * c + 2 * half];
      accs = __builtin_amdgcn_wmma_f32_16x16x4_f32(
          false, a, false, qv[c], (short)0, accs, false, false);
    }

    // ---- masked / scaled scores + online softmax ----
    float scr[8], e[8], vsc[8];
    float tmax = -__builtin_inff();
#pragma unroll
    for (int r = 0; r < 8; ++r) {
      const int s = s0 + 8 * half + r;  // VGPR r <-> row r (+8 for high half)
      const float ksc = k_scaler[(size_t)b * S_ + s];
      vsc[r] = v_scaler[(size_t)b * S_ + s];
      const bool valid = wrap ? (s >= st || s < en) : (s >= st && s < en);
      scr[r] = accs[r] * RSQRT_D * ksc + (valid ? 0.0f : MASK_VALUE);
      tmax = fmaxf(tmax, scr[r]);
    }
    tmax = fmaxf(tmax, __shfl_xor(tmax, 16, 32));
    const float m_new = fmaxf(m_run, tmax);
    const float alpha = __expf(m_run - m_new);
    float tsum = 0.0f;
#pragma unroll
    for (int r = 0; r < 8; ++r) { e[r] = __expf(scr[r] - m_new); tsum += e[r]; }
    tsum += __shfl_xor(tsum, 16, 32);
    l_run = l_run * alpha + tsum;
    m_run = m_new;
#pragma unroll
    for (int j = 0; j < 8; ++j) acc_o[j] *= alpha;

    // publish v_scaler-weighted probabilities for this tile (wave-local)
    if (l16 == 0) {
#pragma unroll
      for (int r = 0; r < 8; ++r) lP[8 * half + r] = e[r] * vsc[r];
    }
    asm volatile("s_wait_dscnt 0" ::: "memory");  // wave-lockstep visibility

    // ---- out += P(1x16) . V_tile(16x128): 8 col-tiles x 4 K-chunks = 32 WMMA
    // A: p broadcast over the 16 M rows; B: V columns 16j..16j+15.
#pragma unroll
    for (int j = 0; j < 8; ++j) {
#pragma unroll
      for (int kk = 0; kk < 4; ++kk) {
        v2f a, bv;
        a.x = lP[4 * kk + 2 * half];
        a.y = lP[4 * kk + 2 * half + 1];
        bv.x = lV[buf][(4 * kk + 2 * half) * LDK + 16 * j + l16];
        bv.y = lV[buf][(4 * kk + 2 * half + 1) * LDK + 16 * j + l16];
        acc_o[j] = __builtin_amdgcn_wmma_f32_16x16x4_f32(
            false, a, false, bv, (short)0, acc_o[j], false, false);
      }
    }
  }

  // Partial results: every D row of acc_o is identical (A was row-broadcast),
  // lanes 0-15 hold columns 16j+l16.
  float* wacc = ws_acc + (size_t)item * D_;
  if (half == 0) {
#pragma unroll
    for (int j = 0; j < 8; ++j) wacc[16 * j + l16] = acc_o[j][0];
  }
  if (lane == 0) { ws_m[item] = m_run; ws_l[item] = l_run; }
}

// ---------------------------------------------------------------------------
// Pass 2: merge the NSPLIT chunks per (b,h) with the log-sum-exp combine.
// ---------------------------------------------------------------------------
__global__ __launch_bounds__(128) void attn_reduce(
    const float* __restrict__ ws_acc, const float* __restrict__ ws_m,
    const float* __restrict__ ws_l,
    float* __restrict__ out, float* __restrict__ m_out,
    float* __restrict__ l_out)
{
  const int bh = blockIdx.x;   // 0..255
  const int d  = threadIdx.x;  // 0..127

  float gm = -__builtin_inff();
#pragma unroll
  for (int c = 0; c < NSPLIT; ++c)
    gm = fmaxf(gm, ws_m[bh * NSPLIT + c]);

  float l = 0.0f, acc = 0.0f;
#pragma unroll
  for (int c = 0; c < NSPLIT; ++c) {
    const float wgt = __expf(ws_m[bh * NSPLIT + c] - gm);
    l += ws_l[bh * NSPLIT + c] * wgt;
    acc += ws_acc[((size_t)bh * NSPLIT + c) * D_ + d] * wgt;
  }
  const float lsafe = (l == 0.0f) ? 1.0f : l;
  out[(size_t)bh * D_ + d] = acc / lsafe;
  if (d == 0) { m_out[bh] = gm; l_out[bh] = lsafe; }
}

// ---------------------------------------------------------------------------
extern "C" void kernel_launch(void* const* d_in, const int* in_sizes, int n_in,
                              void* d_out, int out_size, void* d_ws,
                              size_t ws_size, hipStream_t stream) {
  (void)in_sizes; (void)n_in; (void)out_size; (void)ws_size;
  const float* q   = (const float*)d_in[0];
  const float* k   = (const float*)d_in[1];
  const float* v   = (const float*)d_in[2];
  const float* ksc = (const float*)d_in[3];
  const float* vsc = (const float*)d_in[4];
  const int* start = (const int*)d_in[5];
  const int* end   = (const int*)d_in[6];

  // d_out = [out (B*H*D), m (B*H), l (B*H)] flat in return order
  float* out   = (float*)d_out;
  float* m_out = out + (size_t)B_ * H_ * D_;
  float* l_out = m_out + (size_t)B_ * H_;

  // workspace: acc[B*H*NSPLIT][128] + m + l  (~2.03 MB)
  float* ws_acc = (float*)d_ws;
  float* ws_m   = ws_acc + (size_t)B_ * H_ * NSPLIT * D_;
  float* ws_l   = ws_m + (size_t)B_ * H_ * NSPLIT;

  const int items = B_ * H_ * NSPLIT;           // 4096 waves
  attn_partial<<<dim3(items), dim3(32), 0, stream>>>(
      q, k, v, ksc, vsc, start, end, ws_acc, ws_m, ws_l);
  attn_reduce<<<dim3(B_ * H_), dim3(128), 0, stream>>>(
      ws_acc, ws_m, ws_l, out, m_out, l_out);
}